// MultiBoxLoss_72670846648692
// MI455X (gfx1250) — compile-verified
//
#include <hip/hip_runtime.h>
#include <hip/hip_bf16.h>

#define NB 64
#define NP 8732
#define NO 16
#define NC 81
#define IOU_TH 0.5f

typedef __attribute__((ext_vector_type(2))) float v2f;
typedef __attribute__((ext_vector_type(8))) float v8f;

// ---------------------------------------------------------------- init
__global__ void mbl_init_kernel(unsigned long long* best_key, int* n_pos, float* acc) {
    int i = blockIdx.x * blockDim.x + threadIdx.x;
    if (i < NB * NO) best_key[i] = 0ull;
    if (i < NB)      n_pos[i] = 0;
    if (i < 3)       acc[i] = 0.0f;   // [0]=conf_pos [1]=loc [2]=hard_neg
}

// ------------------------------------------------- match priors <-> GT
__global__ void mbl_match_kernel(const float* __restrict__ gt_boxes,
                                 const float* __restrict__ priors,
                                 float* __restrict__ best_iou,
                                 int* __restrict__ best_gt,
                                 unsigned long long* __restrict__ best_key) {
    __shared__ unsigned long long skey[NO];
    const int b = blockIdx.y;
    const int p = blockIdx.x * blockDim.x + threadIdx.x;
    if (threadIdx.x < NO) skey[threadIdx.x] = 0ull;
    __syncthreads();
    if (p < NP) {
        const float* pr = priors + p * 4;
        const float px1 = pr[0] - pr[2] * 0.5f, py1 = pr[1] - pr[3] * 0.5f;
        const float px2 = pr[0] + pr[2] * 0.5f, py2 = pr[1] + pr[3] * 0.5f;
        const float pa = (px2 - px1) * (py2 - py1);
        float bi = -1.0f; int bg = 0;
        for (int o = 0; o < NO; ++o) {
            const float* gb = gt_boxes + (b * NO + o) * 4;
            const float gx1 = gb[0], gy1 = gb[1], gx2 = gb[2], gy2 = gb[3];
            const float ga = (gx2 - gx1) * (gy2 - gy1);
            float ix = fminf(px2, gx2) - fmaxf(px1, gx1); ix = fmaxf(ix, 0.0f);
            float iy = fminf(py2, gy2) - fmaxf(py1, gy1); iy = fmaxf(iy, 0.0f);
            const float inter = ix * iy;
            const float iou = inter / (pa + ga - inter);
            if (iou > bi) { bi = iou; bg = o; }   // first-max wins (ascending o)
            // iou >= 0 so raw float bits are order-preserving; ~p breaks ties
            // toward the smallest prior index (matches jnp.argmax axis=0).
            unsigned long long key =
                ((unsigned long long)__float_as_uint(iou) << 32) |
                (unsigned long long)(0xFFFFFFFFu - (unsigned)p);
            atomicMax(&skey[o], key);
        }
        best_iou[(long)b * NP + p] = bi;
        best_gt[(long)b * NP + p]  = bg;
    }
    __syncthreads();
    if (threadIdx.x < NO)
        atomicMax(&best_key[b * NO + threadIdx.x], skey[threadIdx.x]);
}

// ------------------------------------------- force-match best prior per GT
__global__ void mbl_apply_kernel(const unsigned long long* __restrict__ best_key,
                                 float* __restrict__ best_iou,
                                 int* __restrict__ best_gt) {
    const int b = threadIdx.x;
    if (b >= NB) return;
    for (int o = 0; o < NO; ++o) {
        unsigned pidx = 0xFFFFFFFFu - (unsigned)(best_key[b * NO + o] & 0xFFFFFFFFull);
        best_iou[(long)b * NP + pidx] = 2.0f;
        best_gt[(long)b * NP + pidx]  = o;
    }
}

// --------------------- loc loss + labels + n_pos (WMMA wave reduction)
__global__ void mbl_loc_kernel(const float* __restrict__ pred_locs,
                               const float* __restrict__ gt_boxes,
                               const int* __restrict__ gt_labels,
                               const float* __restrict__ priors,
                               const float* __restrict__ best_iou,
                               int* __restrict__ best_gt,   // rewritten -> label
                               int* __restrict__ n_pos,
                               float* __restrict__ acc) {
    const int b = blockIdx.y;
    const int p = blockIdx.x * blockDim.x + threadIdx.x;
    const bool active = (p < NP);
    const int pc = active ? p : NP - 1;
    const long idx = (long)b * NP + pc;

    const float iou = best_iou[idx];
    const int g = best_gt[idx];
    const int lbl = (iou < IOU_TH) ? 0 : gt_labels[b * NO + g];
    const bool pos = active && (lbl > 0);

    float locv = 0.0f;
    if (pos) {
        const float* gb = gt_boxes + (b * NO + g) * 4;
        const float cx = (gb[0] + gb[2]) * 0.5f, cy = (gb[1] + gb[3]) * 0.5f;
        const float w = gb[2] - gb[0], h = gb[3] - gb[1];
        const float* pr = priors + pc * 4;
        float t[4];
        t[0] = (cx - pr[0]) / pr[2] * 10.0f;
        t[1] = (cy - pr[1]) / pr[3] * 10.0f;
        t[2] = logf(w / pr[2]) * 5.0f;
        t[3] = logf(h / pr[3]) * 5.0f;
        const float* pl = pred_locs + idx * 4;
        #pragma unroll
        for (int i = 0; i < 4; ++i) {
            const float d = fabsf(pl[i] - t[i]);
            locv += (d < 1.0f) ? 0.5f * d * d : (d - 0.5f);
        }
    }
    if (active) best_gt[idx] = lbl;   // in-place: gt index -> class label

    // --- wave-wide sum via V_WMMA_F32_16X16X4_F32 with B = ones:
    // D[m][n] = rowsum(A,m), so sum of all 256 D entries = 16 * sum(A).
    // EXEC is all-ones here (straight-line, full 256-thread blocks).
    v2f a;  a.x = locv; a.y = 0.0f;
    v2f bo; bo.x = 1.0f; bo.y = 1.0f;
    v8f c = {};
    v8f d = __builtin_amdgcn_wmma_f32_16x16x4_f32(false, a, false, bo,
                                                  (short)0, c, false, false);
    float s = d[0] + d[1] + d[2] + d[3] + d[4] + d[5] + d[6] + d[7];
    #pragma unroll
    for (int m = 16; m >= 1; m >>= 1) s += __shfl_xor(s, m, 32);
    s *= (1.0f / 16.0f);              // each row-sum counted 16x

    int cnt = pos ? 1 : 0;
    #pragma unroll
    for (int m = 16; m >= 1; m >>= 1) cnt += __shfl_xor(cnt, m, 32);

    if ((threadIdx.x & 31) == 0) {
        if (s != 0.0f) atomicAdd(&acc[1], s);
        if (cnt)       atomicAdd(&n_pos[b], cnt);
    }
}

// ------------------------------- cross entropy: one wave32 per prior
__global__ void mbl_ce_kernel(const float* __restrict__ pred_scores,
                              const int* __restrict__ labels,
                              float* __restrict__ ce_neg,
                              float* __restrict__ acc) {
    const int b = blockIdx.y;
    const int wave = threadIdx.x >> 5;
    const int lane = threadIdx.x & 31;
    const int p = blockIdx.x * 8 + wave;
    const bool active = (p < NP);
    const int pc = active ? p : NP - 1;
    const float* row = pred_scores + ((long)b * NP + pc) * NC;

    const bool v1 = (lane + 32) < NC, v2 = (lane + 64) < NC;
    const float x0 = row[lane];
    const float x1 = v1 ? row[lane + 32] : -INFINITY;
    const float x2 = v2 ? row[lane + 64] : -INFINITY;
    float m = fmaxf(x0, fmaxf(x1, x2));
    #pragma unroll
    for (int k = 16; k >= 1; k >>= 1) m = fmaxf(m, __shfl_xor(m, k, 32));
    float s = expf(x0 - m) + (v1 ? expf(x1 - m) : 0.0f) + (v2 ? expf(x2 - m) : 0.0f);
    #pragma unroll
    for (int k = 16; k >= 1; k >>= 1) s += __shfl_xor(s, k, 32);

    const int lbl = labels[(long)b * NP + pc];
    const float ce = logf(s) + m - row[lbl];

    if (active && lane == 0) {
        if (lbl > 0) { atomicAdd(&acc[0], ce); ce_neg[(long)b * NP + p] = 0.0f; }
        else         { ce_neg[(long)b * NP + p] = ce; }
    }
}

// --------- hard-negative mining: exact top-k sum via bitwise threshold
__global__ void mbl_hardneg_kernel(const float* __restrict__ ce_neg,
                                   const int* __restrict__ n_pos,
                                   float* __restrict__ acc) {
    __shared__ float buf[NP];
    __shared__ int s_cnt;
    __shared__ float s_sum;
    const int b = blockIdx.x;
    const int t = threadIdx.x;
    for (int i = t; i < NP; i += blockDim.x) buf[i] = ce_neg[(long)b * NP + i];
    int k = n_pos[b] * 3;
    if (k > NP) k = NP;
    __syncthreads();
    if (k <= 0) return;   // uniform per block

    // k-th largest value's bits (values are all >= 0 so bits are monotone)
    unsigned u = 0;
    for (int bit = 31; bit >= 0; --bit) {
        const unsigned cand = u | (1u << bit);
        if (t == 0) s_cnt = 0;
        __syncthreads();
        int c = 0;
        for (int i = t; i < NP; i += blockDim.x)
            c += (__float_as_uint(buf[i]) >= cand) ? 1 : 0;
        atomicAdd(&s_cnt, c);
        __syncthreads();
        const int cnt = s_cnt;
        __syncthreads();
        if (cnt >= k) u = cand;
    }
    if (t == 0) { s_cnt = 0; s_sum = 0.0f; }
    __syncthreads();
    int cg = 0; float sg = 0.0f;
    for (int i = t; i < NP; i += blockDim.x) {
        const unsigned vb = __float_as_uint(buf[i]);
        if (vb > u) { ++cg; sg += buf[i]; }
    }
    atomicAdd(&s_cnt, cg);
    atomicAdd(&s_sum, sg);
    __syncthreads();
    if (t == 0) {
        const float total = s_sum + (float)(k - s_cnt) * __uint_as_float(u);
        atomicAdd(&acc[2], total);
    }
}

// ---------------------------------------------------------- finalize
__global__ void mbl_final_kernel(const int* __restrict__ n_pos,
                                 const float* __restrict__ acc,
                                 float* __restrict__ out) {
    int total = 0;
    for (int b = 0; b < NB; ++b) total += n_pos[b];
    const float denom = fmaxf((float)total, 1.0f);
    const float loss = (acc[0] + acc[2] + acc[1]) / denom;
    out[0] = (total == 0) ? 0.0f : loss;
}

extern "C" void kernel_launch(void* const* d_in, const int* in_sizes, int n_in,
                              void* d_out, int out_size, void* d_ws, size_t ws_size,
                              hipStream_t stream) {
    const float* pred_locs   = (const float*)d_in[0];
    const float* pred_scores = (const float*)d_in[1];
    const float* gt_boxes    = (const float*)d_in[2];
    const int*   gt_labels   = (const int*)d_in[3];
    const float* priors      = (const float*)d_in[4];
    float* out = (float*)d_out;

    char* ws = (char*)d_ws;
    unsigned long long* best_key = (unsigned long long*)ws;            // NB*NO
    float* best_iou = (float*)(ws + sizeof(unsigned long long) * NB * NO);
    int*   best_gt  = (int*)((char*)best_iou + sizeof(float) * (long)NB * NP);
    float* ce_neg   = (float*)((char*)best_gt + sizeof(int) * (long)NB * NP);
    int*   n_pos    = (int*)((char*)ce_neg + sizeof(float) * (long)NB * NP);
    float* acc      = (float*)(n_pos + NB);

    dim3 blkP((NP + 255) / 256, NB);

    mbl_init_kernel<<<(NB * NO + 255) / 256, 256, 0, stream>>>(best_key, n_pos, acc);
    mbl_match_kernel<<<blkP, 256, 0, stream>>>(gt_boxes, priors, best_iou, best_gt, best_key);
    mbl_apply_kernel<<<1, NB, 0, stream>>>(best_key, best_iou, best_gt);
    mbl_loc_kernel<<<blkP, 256, 0, stream>>>(pred_locs, gt_boxes, gt_labels, priors,
                                             best_iou, best_gt, n_pos, acc);
    mbl_ce_kernel<<<dim3((NP + 7) / 8, NB), 256, 0, stream>>>(pred_scores, best_gt, ce_neg, acc);
    mbl_hardneg_kernel<<<NB, 256, 0, stream>>>(ce_neg, n_pos, acc);
    mbl_final_kernel<<<1, 1, 0, stream>>>(n_pos, acc, out);
}